// SGRUCell_28802050687555
// MI455X (gfx1250) — compile-verified
//
#include <hip/hip_runtime.h>
#include <hip/hip_bf16.h>
#include <math.h>

#define HID 512
#define RR  64
#define DD  256
#define BB  8
#define TT  16
#define G4  2048   // 4*H

typedef __attribute__((ext_vector_type(2))) float v2f;
typedef __attribute__((ext_vector_type(8))) float v8f;

// ---- output offsets (floats), in reference return order ----
static constexpr size_t VF_OFF   = 0;
static constexpr size_t HF_OFF   = 4096;
static constexpr size_t DUF_OFF  = 8192;
static constexpr size_t TROF_OFF = 2105344;
static constexpr size_t TRRF_OFF = 2109440;
static constexpr size_t TREF_OFF = 2113536;
static constexpr size_t HS_OFF   = 4210688;
static constexpr size_t DUS_OFF  = 4276224;
static constexpr size_t MODS_OFF = 37830656;
static constexpr size_t SS_OFF   = 37847040;
static constexpr size_t MS_OFF   = 37847168;
static constexpr size_t RS_OFF   = 37847296;
static constexpr size_t OS_OFF   = 37912832;

__device__ __forceinline__ float sigmoidf_(float x) { return 1.0f / (1.0f + expf(-x)); }
__device__ __forceinline__ float softplusf_(float x) { return (x > 20.f) ? x : log1pf(expf(x)); }

// ---------------------------------------------------------------------------
// Wx GEMM: out[m,n] = sum_k x[m,k]*w[n,k] + bias[n]; M=128, N=2048, K=256.
// One wave per 16x16 tile; V_WMMA_F32_16X16X4_F32 (exact f32 arithmetic).
// ---------------------------------------------------------------------------
__global__ void __launch_bounds__(128) wx_gemm(const float* __restrict__ x,
                                               const float* __restrict__ w,
                                               const float* __restrict__ bias,
                                               float* __restrict__ out) {
    const int wid  = (blockIdx.x * blockDim.x + threadIdx.x) >> 5;
    const int lane = threadIdx.x & 31;
    const int tm = wid >> 7;     // 0..7
    const int tn = wid & 127;    // 0..127
    const int row = tm * 16 + (lane & 15);
    const int col = tn * 16 + (lane & 15);
    const int kh  = (lane < 16) ? 0 : 2;
    const float* xr = x + (size_t)row * DD + kh;
    const float* wr = w + (size_t)col * DD + kh;

    v8f acc = {0.f, 0.f, 0.f, 0.f, 0.f, 0.f, 0.f, 0.f};
#pragma unroll 4
    for (int k = 0; k < DD; k += 4) {
        v2f a, b;
        a.x = xr[k];  a.y = xr[k + 1];
        b.x = wr[k];  b.y = wr[k + 1];
        acc = __builtin_amdgcn_wmma_f32_16x16x4_f32(false, a, false, b,
                                                    (short)0, acc, false, false);
    }
    const int m0 = tm * 16 + ((lane < 16) ? 0 : 8);
    const int n  = tn * 16 + (lane & 15);
    const float bn = bias[n];
#pragma unroll
    for (int r = 0; r < 8; ++r)
        out[(size_t)(m0 + r) * G4 + n] = acc[r] + bn;
}

// ---------------------------------------------------------------------------
// Wh GEMM per step: outp[b,n] = sum_k hpad[b,k]*w[n,k] + bias[n].
// hpad is (16,512) with rows 8..15 kept at zero; M-tile = 16, N=2048, K=512.
// 128 waves; rows 0..7 (held by lanes 0-15) are stored.
// ---------------------------------------------------------------------------
__global__ void __launch_bounds__(128) wh_gemm(const float* __restrict__ hpad,
                                               const float* __restrict__ w,
                                               const float* __restrict__ bias,
                                               float* __restrict__ outp) {
    const int wid  = (blockIdx.x * blockDim.x + threadIdx.x) >> 5;  // 0..127
    const int lane = threadIdx.x & 31;
    const int row = lane & 15;
    const int col = wid * 16 + (lane & 15);
    const int kh  = (lane < 16) ? 0 : 2;
    const float* ar = hpad + (size_t)row * HID + kh;
    const float* wr = w + (size_t)col * HID + kh;

    v8f acc = {0.f, 0.f, 0.f, 0.f, 0.f, 0.f, 0.f, 0.f};
#pragma unroll 4
    for (int k = 0; k < HID; k += 4) {
        v2f a, b;
        a.x = ar[k];  a.y = ar[k + 1];
        b.x = wr[k];  b.y = wr[k + 1];
        acc = __builtin_amdgcn_wmma_f32_16x16x4_f32(false, a, false, b,
                                                    (short)0, acc, false, false);
    }
    if (lane < 16) {  // these lanes hold M = 0..7 (the real batch rows)
        const int n = wid * 16 + lane;
        const float bn = bias[n];
#pragma unroll
        for (int r = 0; r < 8; ++r)
            outp[(size_t)r * G4 + n] = acc[r] + bn;
    }
}

// ---------------------------------------------------------------------------
// fast_add: wh[b, 3H+i] += softplus(alpha) * sum_j dU[b,i,j]*h[b,j].
// One wave per (b,i): 4096 waves spread across the device; lane-split K dot.
// ---------------------------------------------------------------------------
__global__ void __launch_bounds__(256) fast_add(const float* __restrict__ dU,
                                                const float* __restrict__ hbuf,
                                                const float* __restrict__ alpha,
                                                float* __restrict__ wh) {
    const int wid  = (blockIdx.x * 256 + threadIdx.x) >> 5;  // 0..4095
    const int lane = threadIdx.x & 31;
    const int b = wid >> 9, i = wid & 511;
    const float* dr = dU + ((size_t)b * HID + i) * HID;
    const float* hr = hbuf + (size_t)b * HID;
    float acc = 0.f;
    for (int k = lane; k < HID; k += 32) acc = fmaf(dr[k], hr[k], acc);
#pragma unroll
    for (int off = 16; off > 0; off >>= 1) acc += __shfl_xor(acc, off, 32);
    if (lane == 0) {
        const float spa = softplusf_(alpha[0]);
        wh[(size_t)b * G4 + 3 * HID + i] += spa * acc;
    }
}

// ---------------------------------------------------------------------------
// In-place LayerNorm over rows of length 2048 (one block / row).
// ---------------------------------------------------------------------------
__global__ void __launch_bounds__(256) ln_inplace(float* __restrict__ data,
                                                  const float* __restrict__ g,
                                                  const float* __restrict__ b) {
    __shared__ float red[256], red2[256];
    const int row = blockIdx.x, tid = threadIdx.x;
    float* d = data + (size_t)row * G4;
    float s1 = 0.f, s2 = 0.f;
    for (int n = tid; n < G4; n += 256) { float v = d[n]; s1 += v; s2 += v * v; }
    red[tid] = s1; red2[tid] = s2; __syncthreads();
    for (int st = 128; st > 0; st >>= 1) {
        if (tid < st) { red[tid] += red[tid + st]; red2[tid] += red2[tid + st]; }
        __syncthreads();
    }
    const float mu   = red[0] * (1.f / G4);
    const float var  = red2[0] * (1.f / G4) - mu * mu;
    const float rstd = rsqrtf(var + 1e-5f);
    for (int n = tid; n < G4; n += 256)
        d[n] = (d[n] - mu) * rstd * g[n] + b[n];
}

// ---------------------------------------------------------------------------
// zero rows 8..15 of both (16,512) h ping-pong buffers (once at init).
// ---------------------------------------------------------------------------
__global__ void zero_hpad(float* __restrict__ wsH) {
    const int idx = blockIdx.x * 256 + threadIdx.x;   // 0..8191
    const int buf = idx >> 12, off = idx & 4095;
    wsH[buf * 8192 + 4096 + off] = 0.f;
}

// ---------------------------------------------------------------------------
// stepA2: per-batch tail: LN(Wh) + Wx -> gates -> v/h/traces -> mod -> s,m.
// ---------------------------------------------------------------------------
__global__ void __launch_bounds__(512) stepA2(
    const float* __restrict__ WxLN, const float* __restrict__ Wh,
    const float* __restrict__ v_in,
    const float* __restrict__ tro_in, const float* __restrict__ trr_in,
    float* __restrict__ h_out, float* __restrict__ v_out,
    float* __restrict__ tro_out, float* __restrict__ trr_out,
    const float* __restrict__ lnh_g, const float* __restrict__ lnh_b,
    const float* __restrict__ h2mod_w, const float* __restrict__ h2mod_b,
    const float* __restrict__ mod2hs_w, const float* __restrict__ mod2hs_b,
    const float* __restrict__ mod2hm_w, const float* __restrict__ mod2hm_b,
    float* __restrict__ out, int t)
{
    __shared__ float sh_pre[G4];
    __shared__ float sh_hn[HID];
    __shared__ float sh_mod[2 * RR];
    __shared__ float red[512], red2[512];
    __shared__ float sh_mu, sh_rstd;

    const int b = blockIdx.x;
    const int tid = threadIdx.x;

    float s1 = 0.f, s2 = 0.f;
#pragma unroll
    for (int c = 0; c < 4; ++c) {
        int n = tid + c * 512;
        float v = Wh[(size_t)b * G4 + n];
        sh_pre[n] = v; s1 += v; s2 += v * v;
    }
    red[tid] = s1; red2[tid] = s2; __syncthreads();
    for (int st = 256; st > 0; st >>= 1) {
        if (tid < st) { red[tid] += red[tid + st]; red2[tid] += red2[tid + st]; }
        __syncthreads();
    }
    if (tid == 0) {
        float mu  = red[0] * (1.f / G4);
        float var = red2[0] * (1.f / G4) - mu * mu;
        sh_mu = mu; sh_rstd = rsqrtf(var + 1e-5f);
    }
    __syncthreads();
    const float mu = sh_mu, rstd = sh_rstd;
    const float* wx = WxLN + (size_t)(t * BB + b) * G4;
#pragma unroll
    for (int c = 0; c < 4; ++c) {
        int n = tid + c * 512;
        sh_pre[n] = (sh_pre[n] - mu) * rstd * lnh_g[n] + lnh_b[n] + wx[n];
    }
    __syncthreads();

    {
        const int i = tid;
        float z  = sigmoidf_(sh_pre[i]);
        float o  = sigmoidf_(sh_pre[HID + i]);
        float r  = sigmoidf_(sh_pre[2 * HID + i]);
        float dv = sh_pre[3 * HID + i];
        float vo = v_in[b * HID + i];
        float vn = (1.f - z) * vo + z * dv;
        float hn = fmaxf(vn, 0.f);
        sh_hn[i] = hn;
        v_out[b * HID + i] = vn;
        h_out[b * HID + i] = hn;   // rows 0..7 of padded buffer
        float to = tro_in[b * HID + i], tr = trr_in[b * HID + i];
        tro_out[b * HID + i] = (1.f - o) * to + o * hn;
        trr_out[b * HID + i] = (1.f - r) * tr + r * hn;
        size_t ro = (size_t)(t * BB + b) * HID + i;
        out[HS_OFF + ro] = hn;
        out[RS_OFF + ro] = r;
        out[OS_OFF + ro] = o;
    }
    __syncthreads();

    if (tid < 2 * RR) {
        const float* mw = h2mod_w + (size_t)tid * HID;
        float acc = h2mod_b[tid];
        for (int k = 0; k < HID; ++k) acc = fmaf(mw[k], sh_hn[k], acc);
        float mo = fmaxf(acc, 0.f);
        sh_mod[tid] = mo;
        out[MODS_OFF + (size_t)(t * BB + b) * (2 * RR) + tid] = mo;
    }
    __syncthreads();

    if (tid == 0) {
        float sl = mod2hs_b[0], ml = mod2hm_b[0];
        for (int k = 0; k < RR; ++k) {
            sl = fmaf(sh_mod[k],      mod2hs_w[k], sl);
            ml = fmaf(sh_mod[RR + k], mod2hm_w[k], ml);
        }
        out[SS_OFF + t * BB + b] = sigmoidf_(sl);
        out[MS_OFF + t * BB + b] = ml - tanhf(ml);
    }
}

// ---------------------------------------------------------------------------
// stepB: streaming pass over (B,H,H)=2M elements, float4 vectorized.
// dU_old comes from dUs[t-1] (or the input at t=0); only trE + dUs[t] stored.
// ---------------------------------------------------------------------------
__global__ void __launch_bounds__(256) stepB(
    const float* __restrict__ dU_in, float* __restrict__ trE,
    const float* __restrict__ h_new, const float* __restrict__ tro_old,
    const float* __restrict__ trr_old,
    const float* __restrict__ h2h_w,
    const float* __restrict__ alpha, const float* __restrict__ tau_U,
    const float* __restrict__ s_arr, const float* __restrict__ m_arr,
    float* __restrict__ dUs)
{
    const size_t e = ((size_t)blockIdx.x * 256 + threadIdx.x) * 4;
    const int b   = (int)(e >> 18);
    const int rem = (int)(e & 262143);
    const int i   = rem >> 9;
    const int j   = rem & 511;

    const float spa  = softplusf_(alpha[0]) + 1e-8f;
    const float ispa = 1.f / spa;
    const float tau  = sigmoidf_(tau_U[0]);
    const float s = s_arr[b], m = m_arr[b];

    const float hi  = h_new[b * HID + i];
    const float ui  = (tro_old[b * HID + i] + 1.f) * hi;
    const float tri = trr_old[b * HID + i];

    const float4 trj = *reinterpret_cast<const float4*>(trr_old + b * HID + j);
    const float4 hj  = *reinterpret_cast<const float4*>(h_new   + b * HID + j);
    const float4 toj = *reinterpret_cast<const float4*>(tro_old + b * HID + j);
    float4 E4 = *reinterpret_cast<const float4*>(trE + e);
    float4 U4 = *reinterpret_cast<const float4*>(dU_in + e);
    const float4 W4 = *reinterpret_cast<const float4*>(
        h2h_w + (size_t)(3 * HID + i) * HID + j);

#define SGRU_COMP(c)                                            \
    {                                                           \
        float uj  = (toj.c + 1.f) * hj.c;                       \
        float A   = ui * trj.c - tri * uj;                      \
        float En  = (1.f - s) * E4.c + s * A;                   \
        float w_  = W4.c;                                       \
        float up  = fmaxf(1.f - w_, 0.f) * ispa;                \
        float lo  = -fmaxf(1.f + w_, 0.f) * ispa;               \
        float Un  = (1.f - tau) * U4.c + tau * m * En;          \
        Un = fminf(Un, up); Un = fmaxf(Un, lo);                 \
        E4.c = En; U4.c = Un;                                   \
    }
    SGRU_COMP(x) SGRU_COMP(y) SGRU_COMP(z) SGRU_COMP(w)
#undef SGRU_COMP

    *reinterpret_cast<float4*>(trE + e) = E4;
    *reinterpret_cast<float4*>(dUs + e) = U4;
}

// ---------------------------------------------------------------------------
extern "C" void kernel_launch(void* const* d_in, const int* in_sizes, int n_in,
                              void* d_out, int out_size, void* d_ws, size_t ws_size,
                              hipStream_t stream) {
    (void)in_sizes; (void)n_in; (void)out_size; (void)ws_size;

    const float* x        = (const float*)d_in[0];
    const float* dU0      = (const float*)d_in[3];
    const float* x2h_w    = (const float*)d_in[7];
    const float* x2h_b    = (const float*)d_in[8];
    const float* h2h_w    = (const float*)d_in[9];
    const float* h2h_b    = (const float*)d_in[10];
    const float* lnx_g    = (const float*)d_in[11];
    const float* lnx_b    = (const float*)d_in[12];
    const float* lnh_g    = (const float*)d_in[13];
    const float* lnh_b    = (const float*)d_in[14];
    const float* h2mod_w  = (const float*)d_in[15];
    const float* h2mod_b  = (const float*)d_in[16];
    const float* mod2hs_w = (const float*)d_in[17];
    const float* mod2hs_b = (const float*)d_in[18];
    const float* mod2hm_w = (const float*)d_in[19];
    const float* mod2hm_b = (const float*)d_in[20];
    const float* alpha    = (const float*)d_in[21];
    const float* tau_U    = (const float*)d_in[22];

    float* out = (float*)d_out;
    float* ws  = (float*)d_ws;

    // workspace layout (floats)
    float* wsWx = ws;                 // 262144  (T*B, 4H) = LN(x@W.T + b)
    float* wsWh = ws + 262144;        // 16384   (8, 2048) per-step pre-LN Wh
    float* wsH  = ws + 278528;        // 2 x (16,512) padded ping-pong
    float* wsV  = ws + 294912;        // 2 x 4096
    float* wsTO = ws + 303104;        // 2 x 4096
    float* wsTR = ws + 311296;        // 2 x 4096
    float* wsTE = ws + 319488;        // 2097152 (B,H,H) in-place trE

    // ---- init state ----
    zero_hpad<<<32, 256, 0, stream>>>(wsH);
    hipMemcpyAsync(wsH,  d_in[1], (size_t)BB * HID * 4, hipMemcpyDeviceToDevice, stream);
    hipMemcpyAsync(wsV,  d_in[2], (size_t)BB * HID * 4, hipMemcpyDeviceToDevice, stream);
    hipMemcpyAsync(wsTO, d_in[4], (size_t)BB * HID * 4, hipMemcpyDeviceToDevice, stream);
    hipMemcpyAsync(wsTR, d_in[5], (size_t)BB * HID * 4, hipMemcpyDeviceToDevice, stream);
    hipMemcpyAsync(wsTE, d_in[6], (size_t)BB * HID * HID * 4, hipMemcpyDeviceToDevice, stream);

    // ---- precompute Wx for all t (WMMA f32 GEMM), then LN ----
    wx_gemm<<<256, 128, 0, stream>>>(x, x2h_w, x2h_b, wsWx);
    ln_inplace<<<TT * BB, 256, 0, stream>>>(wsWx, lnx_g, lnx_b);

    // ---- recurrent scan ----
    for (int t = 0; t < TT; ++t) {
        const int rb = t & 1, wb = (t + 1) & 1;
        const float* dU_prev =
            (t == 0) ? dU0 : (out + DUS_OFF + (size_t)(t - 1) * BB * HID * HID);

        wh_gemm<<<32, 128, 0, stream>>>(wsH + rb * 8192, h2h_w, h2h_b, wsWh);
        fast_add<<<512, 256, 0, stream>>>(dU_prev, wsH + rb * 8192, alpha, wsWh);
        stepA2<<<BB, 512, 0, stream>>>(
            wsWx, wsWh,
            wsV + rb * 4096, wsTO + rb * 4096, wsTR + rb * 4096,
            wsH + wb * 8192, wsV + wb * 4096, wsTO + wb * 4096, wsTR + wb * 4096,
            lnh_g, lnh_b, h2mod_w, h2mod_b,
            mod2hs_w, mod2hs_b, mod2hm_w, mod2hm_b,
            out, t);
        stepB<<<2048, 256, 0, stream>>>(
            dU_prev, wsTE,
            wsH + wb * 8192, wsTO + rb * 4096, wsTR + rb * 4096,
            h2h_w, alpha, tau_U,
            out + SS_OFF + (size_t)t * BB, out + MS_OFF + (size_t)t * BB,
            out + DUS_OFF + (size_t)t * BB * HID * HID);
    }

    // ---- final carry (T even -> buffer 0) ----
    hipMemcpyAsync(out + VF_OFF,   wsV,  (size_t)BB * HID * 4, hipMemcpyDeviceToDevice, stream);
    hipMemcpyAsync(out + HF_OFF,   wsH,  (size_t)BB * HID * 4, hipMemcpyDeviceToDevice, stream);
    hipMemcpyAsync(out + DUF_OFF,  out + DUS_OFF + (size_t)(TT - 1) * BB * HID * HID,
                   (size_t)BB * HID * HID * 4, hipMemcpyDeviceToDevice, stream);
    hipMemcpyAsync(out + TROF_OFF, wsTO, (size_t)BB * HID * 4, hipMemcpyDeviceToDevice, stream);
    hipMemcpyAsync(out + TRRF_OFF, wsTR, (size_t)BB * HID * 4, hipMemcpyDeviceToDevice, stream);
    hipMemcpyAsync(out + TREF_OFF, wsTE, (size_t)BB * HID * HID * 4, hipMemcpyDeviceToDevice, stream);
}